// SoftDotAttention_19602230739452
// MI455X (gfx1250) — compile-verified
//
#include <hip/hip_runtime.h>

// Problem constants from the reference: B=32, S=8192, D=256, fp32 everywhere.
#define BB 32
#define SS 8192
#define DD 256
#define NEGV (-1e30f)

typedef __attribute__((ext_vector_type(2))) float v2f;
typedef __attribute__((ext_vector_type(8))) float v8f;

__device__ __forceinline__ float wave_sum(float v) {
    #pragma unroll
    for (int off = 16; off > 0; off >>= 1) v += __shfl_xor(v, off, 32);
    return v;
}
__device__ __forceinline__ float wave_max(float v) {
    #pragma unroll
    for (int off = 16; off > 0; off >>= 1) v = fmaxf(v, __shfl_xor(v, off, 32));
    return v;
}

// ---------------------------------------------------------------------------
// Pass 1: attn[b,s] = guard( mask ? NEG : <src[b,s,:], tgt[b,:]> ) * w[b,s]
// One wave handles 32 rows; 32 lanes cover the 1KB row with two float4 loads.
// Writes raw scores into the attn region of d_out.
// ---------------------------------------------------------------------------
__global__ void k_scores(const float* __restrict__ tgt,
                         const float* __restrict__ src,
                         const unsigned char* __restrict__ mask,
                         const float* __restrict__ ew,
                         float* __restrict__ sc) {
    const int wave = threadIdx.x >> 5;
    const int lane = threadIdx.x & 31;
    const int block_row = blockIdx.x * 256;          // global row index in [0, B*S)
    const int b = block_row / SS;                    // S % 256 == 0, no straddle
    const float4 t0 = *(const float4*)(tgt + b * DD + lane * 4);
    const float4 t1 = *(const float4*)(tgt + b * DD + 128 + lane * 4);
    const int row0 = block_row + wave * 32;
    for (int r = 0; r < 32; ++r) {
        const int row = row0 + r;
        const float* p = src + (size_t)row * DD;
        float4 s0 = *(const float4*)(p + lane * 4);
        float4 s1 = *(const float4*)(p + 128 + lane * 4);
        float d = s0.x * t0.x + s0.y * t0.y + s0.z * t0.z + s0.w * t0.w
                + s1.x * t1.x + s1.y * t1.y + s1.z * t1.z + s1.w * t1.w;
        d = wave_sum(d);
        if (lane == 0) {
            float a = mask[row] ? NEGV : d;
            a *= ew[row];
            if (!(a == a) || a <= NEGV) a = NEGV;   // nan_to_num + neg-inf guard
            sc[row] = a;
        }
    }
}

// ---------------------------------------------------------------------------
// Pass 2: in-place softmax over S per batch row. 1024 threads, 8 values each.
// ---------------------------------------------------------------------------
__global__ void k_softmax(float* __restrict__ attn) {
    const int b = blockIdx.x;
    const int t = threadIdx.x;                       // 0..1023
    __shared__ float red[32];
    __shared__ float bcast_max;
    __shared__ float bcast_sum;

    float v[8];
    float m = NEGV;
    #pragma unroll
    for (int i = 0; i < 8; ++i) {
        v[i] = attn[b * SS + t + i * 1024];
        m = fmaxf(m, v[i]);
    }
    m = wave_max(m);
    if ((t & 31) == 0) red[t >> 5] = m;
    __syncthreads();
    if (t < 32) {
        float x = wave_max(red[t]);
        if (t == 0) bcast_max = x;
    }
    __syncthreads();
    const float gm = bcast_max;

    float e[8];
    float s = 0.f;
    #pragma unroll
    for (int i = 0; i < 8; ++i) {
        e[i] = __expf(v[i] - gm);
        s += e[i];
    }
    s = wave_sum(s);
    __syncthreads();                                 // red[] reuse safety
    if ((t & 31) == 0) red[t >> 5] = s;
    __syncthreads();
    if (t < 32) {
        float x = wave_sum(red[t]);
        if (t == 0) bcast_sum = x;
    }
    __syncthreads();
    const float inv = 1.0f / bcast_sum;
    #pragma unroll
    for (int i = 0; i < 8; ++i)
        attn[b * SS + t + i * 1024] = e[i] * inv;
}

// ---------------------------------------------------------------------------
// Pass 3: weighted context via V_WMMA_F32_16X16X4_F32.
//   One wave = (batch b, 16-wide d-tile nt, 512-long s-chunk).
//   A (16x4): attn[s..s+3] replicated across all 16 M rows.
//     ISA layout: VGPR0 = {K0 lanes0-15, K2 lanes16-31}, VGPR1 = {K1, K3}.
//   B (4x16): src[b, s+k, d0..d0+15], rows striped across lanes per VGPR.
//   C (16x16): all rows equal; row 0 -> partial[b, chunk, d0..d0+15].
// ---------------------------------------------------------------------------
__global__ void k_wctx(const float* __restrict__ attn,
                       const float* __restrict__ src,
                       float* __restrict__ part) {
    const int lane = threadIdx.x & 31;
    const int wid  = blockIdx.x * (blockDim.x >> 5) + (threadIdx.x >> 5);
    const int b     = wid >> 8;          // 256 waves per batch
    const int chunk = (wid >> 4) & 15;   // 16 s-chunks of 512
    const int nt    = wid & 15;          // 16 d-tiles of 16
    const int d0 = nt * 16;
    const int s0 = chunk * (SS / 16);
    const int kbase = (lane >> 4) * 2;   // lanes 0-15 -> K{0,1}; 16-31 -> K{2,3}

    const float* ap = attn + b * SS + s0 + kbase;
    const float* bp = src + ((size_t)b * SS + (s0 + kbase)) * DD + d0 + (lane & 15);

    v8f c = {};
    for (int i = 0; i < (SS / 16) / 4; ++i) {        // 128 WMMAs per wave
        v2f a, bm;
        a.x  = ap[0];
        a.y  = ap[1];
        bm.x = bp[0];
        bm.y = bp[DD];
        c = __builtin_amdgcn_wmma_f32_16x16x4_f32(
                /*neg_a=*/false, a, /*neg_b=*/false, bm,
                /*c_mod=*/(short)0, c, /*reuse_a=*/false, /*reuse_b=*/false);
        ap += 4;
        bp += 4 * DD;
    }
    if (lane < 16)
        part[((b * 16 + chunk) << 8) + d0 + lane] = c[0];
}

// ---------------------------------------------------------------------------
// Pass 4: reduce 16 partials per (b,d), leaky ReLU, write h_tilde.
// ---------------------------------------------------------------------------
__global__ void k_final(const float* __restrict__ part, float* __restrict__ h) {
    const int i = blockIdx.x * blockDim.x + threadIdx.x;   // 0..B*D-1
    const int b = i >> 8;
    const int d = i & 255;
    float s = 0.f;
    #pragma unroll
    for (int c = 0; c < 16; ++c) s += part[((b * 16 + c) << 8) + d];
    h[i] = (s >= 0.f) ? s : 0.01f * s;
}

extern "C" void kernel_launch(void* const* d_in, const int* in_sizes, int n_in,
                              void* d_out, int out_size, void* d_ws, size_t ws_size,
                              hipStream_t stream) {
    const float*         tgt  = (const float*)d_in[0];         // (B, D)
    const float*         src  = (const float*)d_in[1];         // (B, S, D)
    const unsigned char* mask = (const unsigned char*)d_in[2]; // (B, S) bool
    const float*         ew   = (const float*)d_in[3];         // (B, S)

    float* out  = (float*)d_out;
    float* h    = out;               // h_tilde: B*D floats
    float* attn = out + BB * DD;     // attn:    B*S floats
    float* part = (float*)d_ws;      // partials: B*16*D floats = 512 KB

    k_scores <<<(BB * SS) / 256, 256, 0, stream>>>(tgt, src, mask, ew, attn);
    k_softmax<<<BB, 1024, 0, stream>>>(attn);
    k_wctx   <<<(BB * 256) / 8, 256, 0, stream>>>(attn, src, part);
    k_final  <<<(BB * DD) / 256, 256, 0, stream>>>(part, h);
}